// org_receiver_61632780698135
// MI455X (gfx1250) — compile-verified
//
#include <hip/hip_runtime.h>

// CDNA5 / gfx1250: wave32, WMMA 16x16x4 FP32.
// Strategy: factor the per-edge MLP (OUT=1) into node-level dense GEMMs
//   Y = X_src @ linW  (WMMA fp32, exact), Yb = X_src @ linb
// then a bandwidth-bound edge pass: msg = dot(Y[src], ea) + Yb[src],
// scattered with global_atomic_add_f32.
// Cache policy: ea (386MB total) is streamed once -> non-temporal loads,
// so the gathered Y tables (<=26MB) and node features (90MB) stay resident
// in the 192MB L2.

typedef __attribute__((ext_vector_type(2))) float v2f;
typedef __attribute__((ext_vector_type(4))) float v4f;
typedef __attribute__((ext_vector_type(8))) float v8f;

#define N_IND 200000
#define N_ORG 100000
#define N_EXT 50000
#define NEDGE 500000
#define FDIM  64
#define NCONV 7

// out[n*7 + col] = dot(x_org[n,:], root[:,0]) + b[0]   (overwrites -> deterministic init)
__global__ void init_col_kernel(const float* __restrict__ x_org,
                                const float* __restrict__ root,
                                const float* __restrict__ b,
                                float* __restrict__ out, int col) {
  int n = blockIdx.x * blockDim.x + threadIdx.x;
  if (n >= N_ORG) return;
  const v4f* xr = (const v4f*)(x_org + (size_t)n * FDIM);
  const v4f* rr = (const v4f*)root;
  float acc = b[0];
#pragma unroll
  for (int j = 0; j < FDIM / 4; ++j) {
    v4f xv = xr[j], rv = rr[j];
    acc += xv.x * rv.x + xv.y * rv.y + xv.z * rv.z + xv.w * rv.w;
  }
  out[(size_t)n * NCONV + col] = acc;
}

// Y[N, ed] = X[N, 64] @ W[64, ed].  N%16==0, ed%16==0, K=64 (16 WMMA steps).
// One wave per 16x16 output tile; 8 waves (256 thr) per block.
// A layout (16x4 f32): lanes 0-15 hold K={0,1}, lanes 16-31 hold K={2,3} (M=lane%16).
// B layout (4x16):     lanes 0-15 hold K={0,1}, lanes 16-31 hold K={2,3} (N=lane%16).
// C/D (16x16 f32):     VGPR r -> M=r (lanes 0-15) / M=r+8 (lanes 16-31), N=lane%16.
__global__ void gemm_wmma_kernel(const float* __restrict__ X,
                                 const float* __restrict__ W,
                                 float* __restrict__ Y, int N, int ed) {
  int wave = threadIdx.x >> 5;
  int lane = threadIdx.x & 31;
  int ntiles = ed >> 4;
  int tile = blockIdx.x * (blockDim.x >> 5) + wave;
  int mt = tile / ntiles;
  if (mt * 16 >= N) return;            // wave-uniform: EXEC stays all-1s at WMMA
  int nt = tile - mt * ntiles;
  int mrow = mt << 4, ncol = nt << 4;
  int l  = lane & 15;
  int kb = (lane >> 4) << 1;           // 0 (low half) or 2 (high half)

  v8f acc = {};
  const float* xrow = X + (size_t)(mrow + l) * FDIM + kb;   // A[M=l, K=kb..kb+1]
  const float* wcol = W + (size_t)kb * ed + ncol + l;       // B[K=kb..kb+1, N=l]
#pragma unroll
  for (int kk = 0; kk < FDIM / 4; ++kk) {
    v2f a, bm;
    a.x  = xrow[0];  a.y  = xrow[1];
    bm.x = wcol[0];  bm.y = wcol[ed];
    acc = __builtin_amdgcn_wmma_f32_16x16x4_f32(false, a, false, bm,
                                                (short)0, acc, false, false);
    xrow += 4;
    wcol += (size_t)4 * ed;
  }
  int mbase = mrow + ((lane >> 4) << 3);   // +8 rows for lanes 16-31
  float* yout = Y + (size_t)mbase * ed + ncol + l;
#pragma unroll
  for (int r = 0; r < 8; ++r) yout[(size_t)r * ed] = acc[r];
}

// Scalar fallback GEMM for tiny ed (conv4: ed=1): Y[n,j] = dot(X[n,:], W[:,j])
__global__ void gemm_small_kernel(const float* __restrict__ X,
                                  const float* __restrict__ W,
                                  float* __restrict__ Y, int N, int ed) {
  int n = blockIdx.x * blockDim.x + threadIdx.x;
  if (n >= N) return;
  const float* xr = X + (size_t)n * FDIM;
  for (int j = 0; j < ed; ++j) {
    float acc = 0.f;
#pragma unroll 16
    for (int i = 0; i < FDIM; ++i) acc += xr[i] * W[(size_t)i * ed + j];
    Y[(size_t)n * ed + j] = acc;
  }
}

// Yb[n] = dot(X[n,:], v[0:64])
__global__ void dotvec_kernel(const float* __restrict__ X,
                              const float* __restrict__ v,
                              float* __restrict__ out, int N) {
  int n = blockIdx.x * blockDim.x + threadIdx.x;
  if (n >= N) return;
  const v4f* xr = (const v4f*)(X + (size_t)n * FDIM);
  const v4f* vr = (const v4f*)v;
  float acc = 0.f;
#pragma unroll
  for (int j = 0; j < FDIM / 4; ++j) {
    v4f xv = xr[j], vv = vr[j];
    acc += xv.x * vv.x + xv.y * vv.y + xv.z * vv.z + xv.w * vv.w;
  }
  out[n] = acc;
}

// msg = dot(Y[src[e]], ea[e]) + Yb[src[e]];  atomic scatter into out column.
// ea is streamed once -> non-temporal loads keep Y hot in L2.
__global__ void edge_kernel(const float* __restrict__ Y,
                            const float* __restrict__ Yb,
                            const float* __restrict__ ea,
                            const int* __restrict__ src,
                            const int* __restrict__ dst,
                            float* __restrict__ out, int col, int ed) {
  int e = blockIdx.x * blockDim.x + threadIdx.x;
  if (e >= NEDGE) return;
  int s = src[e], d = dst[e];
  float acc = Yb[s];
  const float* ye = Y + (size_t)s * ed;
  const float* ae = ea + (size_t)e * ed;
  if ((ed & 3) == 0) {
    for (int j = 0; j < ed; j += 4) {
      v4f yv = *(const v4f*)(ye + j);
      v4f av = __builtin_nontemporal_load((const v4f*)(ae + j));  // TH=NT stream
      acc += yv.x * av.x + yv.y * av.y + yv.z * av.z + yv.w * av.w;
    }
  } else {
    for (int j = 0; j < ed; ++j)
      acc += ye[j] * __builtin_nontemporal_load(ae + j);
  }
  atomicAdd(out + (size_t)d * NCONV + col, acc);   // global_atomic_add_f32 (no return)
}

extern "C" void kernel_launch(void* const* d_in, const int* in_sizes, int n_in,
                              void* d_out, int out_size, void* d_ws, size_t ws_size,
                              hipStream_t stream) {
  (void)in_sizes; (void)n_in; (void)out_size; (void)ws_size;
  const float* x_ind = (const float*)d_in[0];
  const float* x_org = (const float*)d_in[1];
  const float* x_ext = (const float*)d_in[2];
  float* out = (float*)d_out;

  const int    nsrc[NCONV] = {N_IND, N_ORG, N_EXT, N_IND, N_IND, N_ORG, N_EXT};
  const float* xsrc[NCONV] = {x_ind, x_org, x_ext, x_ind, x_ind, x_org, x_ext};
  const int    eds[NCONV]  = {32, 32, 32, 1, 32, 32, 32};

  // Workspace reuse per conv (stream-ordered): Y (<=200000*32 f32) + Yb (<=200000 f32) ~26.4MB
  float* Y  = (float*)d_ws;
  float* Yb = (float*)d_ws + (size_t)N_IND * 32;

  for (int c = 0; c < NCONV; ++c) {
    const int*   srcv = (const int*)  d_in[3 + c * 7 + 0];
    const int*   dstv = (const int*)  d_in[3 + c * 7 + 1];
    const float* ea   = (const float*)d_in[3 + c * 7 + 2];
    const float* lw   = (const float*)d_in[3 + c * 7 + 3];
    const float* lb   = (const float*)d_in[3 + c * 7 + 4];
    const float* rt   = (const float*)d_in[3 + c * 7 + 5];
    const float* bb   = (const float*)d_in[3 + c * 7 + 6];
    int N = nsrc[c], ed = eds[c];

    init_col_kernel<<<(N_ORG + 255) / 256, 256, 0, stream>>>(x_org, rt, bb, out, c);

    if ((ed & 15) == 0) {
      int tiles = (N / 16) * (ed / 16);
      gemm_wmma_kernel<<<(tiles + 7) / 8, 256, 0, stream>>>(xsrc[c], lw, Y, N, ed);
    } else {
      gemm_small_kernel<<<(N + 255) / 256, 256, 0, stream>>>(xsrc[c], lw, Y, N, ed);
    }
    dotvec_kernel<<<(N + 255) / 256, 256, 0, stream>>>(xsrc[c], lb, Yb, N);

    edge_kernel<<<(NEDGE + 255) / 256, 256, 0, stream>>>(Y, Yb, ea, srcv, dstv, out, c, ed);
  }
}